// SimpleBERT_1159641169963
// MI455X (gfx1250) — compile-verified
//
#include <hip/hip_runtime.h>
#include <math.h>

// ---------------------------------------------------------------------------
// SimpleBERT forward for MI455X (gfx1250), fp32 end-to-end.
// All matmuls routed through V_WMMA_F32_16X16X4_F32 (wave32 WMMA).
// Compile-time N/K so inner loops fully unroll with no exec-mask juggling.
// SEQ=512 D=64 H=4 HD=16 FF=128 V=4
// ---------------------------------------------------------------------------

#define SEQ 512
#define DMODEL 64
#define NHEAD 4
#define HDIM 16
#define FFDIM 128
#define VOCAB 4
#define LN_EPS 1e-5f
#define ATT_SCALE 0.25f /* 1/sqrt(16) */

typedef float v2f __attribute__((ext_vector_type(2)));
typedef float v8f __attribute__((ext_vector_type(8)));

__device__ __forceinline__ v8f wmma_f32x4(v2f a, v2f b, v8f c) {
  // 8 args: (neg_a, A, neg_b, B, c_mod, C, reuse_a, reuse_b)
  return __builtin_amdgcn_wmma_f32_16x16x4_f32(false, a, false, b, (short)0, c,
                                               false, false);
}

// x[s,d] = tok_emb[token_ids[s], d] + pos_emb[pos_ids[s], d]
__global__ void embed_kernel(const int* __restrict__ tok,
                             const int* __restrict__ pos,
                             const float* __restrict__ tok_emb,
                             const float* __restrict__ pos_emb,
                             float* __restrict__ x) {
  int i = blockIdx.x * blockDim.x + threadIdx.x;
  if (i >= SEQ * DMODEL) return;
  int s = i >> 6;
  int d = i & (DMODEL - 1);
  x[i] = tok_emb[tok[s] * DMODEL + d] + pos_emb[pos[s] * DMODEL + d];
}

// out[SEQ,N] = A[SEQ,K] @ W[N,K]^T + bias[N], optional exact GELU.
// N, K compile-time (multiples of 16 / 4). One wave per 16x16 output tile.
template <int N, int K, bool GELU>
__global__ void gemm_bias_kernel(const float* __restrict__ A,
                                 const float* __restrict__ W,
                                 const float* __restrict__ bias,
                                 float* __restrict__ out) {
  const int lane = threadIdx.x & 31;
  const int wave = threadIdx.x >> 5;
  constexpr int tilesN = N / 16;
  const int tile = blockIdx.x * (blockDim.x >> 5) + wave;
  const int mt = tile / tilesN;
  const int nt = tile % tilesN;
  const int l15 = lane & 15;
  const int hl = lane >> 4;  // 0: K pair (k0,k0+1), 1: (k0+2,k0+3)
  const int m = (mt << 4) + l15;
  const int n = (nt << 4) + l15;
  const float* __restrict__ Ap = A + m * K + hl * 2;
  const float* __restrict__ Wp = W + n * K + hl * 2;

  v8f c = {};
#pragma unroll
  for (int k0 = 0; k0 < K; k0 += 4) {
    v2f a = *(const v2f*)(Ap + k0);  // global_load_b64
    v2f b = *(const v2f*)(Wp + k0);  // global_load_b64
    c = wmma_f32x4(a, b, c);
  }
  const float bv = bias[n];
#pragma unroll
  for (int p = 0; p < 8; ++p) {
    const int row = (mt << 4) + p + hl * 8;
    float val = c[p] + bv;
    if (GELU) val = 0.5f * val * (1.0f + erff(val * 0.70710678118654752f));
    out[row * N + n] = val;
  }
}

// logits[SEQ,4] = A[SEQ,64] @ Wo[4,64]^T + bo.  The ragged N=4 tile is handled
// by staging a zero-padded 16x64 weight tile in LDS: the WMMA loop stays
// divergence-free; only the 8 final stores are lane-masked.
__global__ void logits_kernel(const float* __restrict__ A,
                              const float* __restrict__ Wo,
                              const float* __restrict__ bo,
                              float* __restrict__ out) {
  __shared__ float wl[16 * DMODEL];
  const int tid = threadIdx.x;  // blockDim = 128
  for (int i = tid; i < 16 * DMODEL; i += 128)
    wl[i] = (i < VOCAB * DMODEL) ? Wo[i] : 0.0f;
  __syncthreads();

  const int lane = tid & 31;
  const int wave = tid >> 5;
  const int l15 = lane & 15;
  const int hl = lane >> 4;
  const int mt = blockIdx.x * 4 + wave;
  const int m = (mt << 4) + l15;
  const float* __restrict__ Ap = A + m * DMODEL + hl * 2;
  const float* __restrict__ Wp = wl + l15 * DMODEL + hl * 2;

  v8f c = {};
#pragma unroll
  for (int k0 = 0; k0 < DMODEL; k0 += 4) {
    v2f a = *(const v2f*)(Ap + k0);  // global_load_b64
    v2f b = *(const v2f*)(Wp + k0);  // ds_load_b64
    c = wmma_f32x4(a, b, c);
  }
  if (l15 < VOCAB) {
    const float bv = bo[l15];
#pragma unroll
    for (int p = 0; p < 8; ++p) {
      const int row = (mt << 4) + p + hl * 8;
      out[row * VOCAB + l15] = c[p] + bv;
    }
  }
}

// One workgroup = one wave = (head h, 16-query block qb).
// Phase 0: stage V head transposed into LDS (vt[hd][t], padded stride)
// Phase 1: scores(16x512) = Q_blk @ K^T via WMMA -> LDS
// Phase 2: wave-cooperative softmax per row (shfl_xor reductions, wave32)
// Phase 3: out(16x16) = P(16x512) @ V_head(512x16), 1/den folded into A reads
#define VS (SEQ + 4) /* padded LDS row stride, keeps 8B alignment, no bank hit */
__global__ void attn_kernel(const float* __restrict__ q,
                            const float* __restrict__ k,
                            const float* __restrict__ v,
                            float* __restrict__ out) {
  __shared__ float sc[16 * SEQ];
  __shared__ float vt[HDIM * VS];
  __shared__ float inv_den[16];
  const int lane = threadIdx.x & 31;
  const int h = blockIdx.x & (NHEAD - 1);
  const int qb = blockIdx.x >> 2;
  const int l15 = lane & 15;
  const int hl = lane >> 4;
  const int m = (qb << 4) + l15;
  const int hoff = h * HDIM;

  // --- stage V head transposed: vt[hd][t] = v[t, hoff+hd] ---
  for (int t = lane; t < SEQ; t += 32) {
    const float* __restrict__ vp = v + t * DMODEL + hoff;
#pragma unroll
    for (int hd = 0; hd < HDIM; ++hd) vt[hd * VS + t] = vp[hd];
  }

  // --- scores = (Q @ K^T) * scale ---
  const float* __restrict__ qp = q + m * DMODEL + hoff + hl * 2;
  for (int t = 0; t < SEQ / 16; ++t) {
    const float* __restrict__ kp = k + ((t << 4) + l15) * DMODEL + hoff + hl * 2;
    v8f c = {};
#pragma unroll
    for (int k0 = 0; k0 < HDIM; k0 += 4) {
      v2f a = *(const v2f*)(qp + k0);
      v2f b = *(const v2f*)(kp + k0);
      c = wmma_f32x4(a, b, c);
    }
#pragma unroll
    for (int p = 0; p < 8; ++p)
      sc[(p + hl * 8) * SEQ + (t << 4) + l15] = c[p] * ATT_SCALE;
  }
  __syncthreads();

  // --- softmax over each of the 16 rows (512 wide), wave cooperative ---
  for (int r = 0; r < 16; ++r) {
    float mx = -3.0e38f;
    for (int i = lane; i < SEQ; i += 32) mx = fmaxf(mx, sc[r * SEQ + i]);
#pragma unroll
    for (int off = 16; off > 0; off >>= 1)
      mx = fmaxf(mx, __shfl_xor(mx, off, 32));
    float sum = 0.0f;
    for (int i = lane; i < SEQ; i += 32) {
      float e = __expf(sc[r * SEQ + i] - mx);
      sc[r * SEQ + i] = e;
      sum += e;
    }
#pragma unroll
    for (int off = 16; off > 0; off >>= 1) sum += __shfl_xor(sum, off, 32);
    if (lane == 0) inv_den[r] = 1.0f / sum;
  }
  __syncthreads();

  // --- out = (P * inv_den_row) @ V_head ---
  const float inv = inv_den[l15];  // A-row owned by this lane
  const float* __restrict__ pp = sc + l15 * SEQ + hl * 2;
  const float* __restrict__ bp = vt + l15 * VS + hl * 2;
  v8f c = {};
#pragma unroll 8
  for (int k0 = 0; k0 < SEQ; k0 += 4) {
    v2f a = *(const v2f*)(pp + k0);  // ds_load_b64
    a.x *= inv;
    a.y *= inv;
    v2f b = *(const v2f*)(bp + k0);  // ds_load_b64 (transposed V)
    c = wmma_f32x4(a, b, c);
  }
#pragma unroll
  for (int p = 0; p < 8; ++p)
    out[((qb << 4) + p + hl * 8) * DMODEL + hoff + l15] = c[p];
}

// out[row,:] = LayerNorm(a[row,:] + b[row,:]) * g + beta   (D=64, 1 wave/row)
__global__ void add_ln_kernel(const float* __restrict__ a,
                              const float* __restrict__ b,
                              const float* __restrict__ g,
                              const float* __restrict__ beta,
                              float* __restrict__ out) {
  const int row = blockIdx.x;
  const int lane = threadIdx.x;  // 32 threads
  const int i0 = row * DMODEL + lane;
  const int i1 = i0 + 32;
  float v0 = a[i0] + b[i0];
  float v1 = a[i1] + b[i1];
  float s = v0 + v1;
#pragma unroll
  for (int off = 16; off > 0; off >>= 1) s += __shfl_xor(s, off, 32);
  const float mu = s * (1.0f / 64.0f);
  const float d0 = v0 - mu, d1 = v1 - mu;
  float vs = d0 * d0 + d1 * d1;
#pragma unroll
  for (int off = 16; off > 0; off >>= 1) vs += __shfl_xor(vs, off, 32);
  const float inv = rsqrtf(vs * (1.0f / 64.0f) + LN_EPS);
  out[i0] = d0 * inv * g[lane] + beta[lane];
  out[i1] = d1 * inv * g[lane + 32] + beta[lane + 32];
}

extern "C" void kernel_launch(void* const* d_in, const int* in_sizes, int n_in,
                              void* d_out, int out_size, void* d_ws,
                              size_t ws_size, hipStream_t stream) {
  (void)in_sizes; (void)n_in; (void)out_size; (void)ws_size;
  const int* tok = (const int*)d_in[0];
  const int* pos = (const int*)d_in[1];
  const float* temb = (const float*)d_in[2];
  const float* pemb = (const float*)d_in[3];
  const float* Wq = (const float*)d_in[4];
  const float* bq = (const float*)d_in[5];
  const float* Wk = (const float*)d_in[6];
  const float* bk = (const float*)d_in[7];
  const float* Wv = (const float*)d_in[8];
  const float* bv = (const float*)d_in[9];
  const float* W1 = (const float*)d_in[10];
  const float* b1 = (const float*)d_in[11];
  const float* W2 = (const float*)d_in[12];
  const float* b2 = (const float*)d_in[13];
  const float* g1 = (const float*)d_in[14];
  const float* be1 = (const float*)d_in[15];
  const float* g2 = (const float*)d_in[16];
  const float* be2 = (const float*)d_in[17];
  const float* Wo = (const float*)d_in[18];
  const float* bo = (const float*)d_in[19];
  float* out = (float*)d_out;

  // workspace layout (fp32), total 327,680 floats = 1.31 MB (lives in L2)
  float* x = (float*)d_ws;         // [512,64] embedded input
  float* q = x + SEQ * DMODEL;     // [512,64]
  float* kk = q + SEQ * DMODEL;    // [512,64]
  float* vv = kk + SEQ * DMODEL;   // [512,64]
  float* ao = vv + SEQ * DMODEL;   // [512,64] attention out
  float* x1 = ao + SEQ * DMODEL;   // [512,64] after LN1
  float* ff = x1 + SEQ * DMODEL;   // [512,128] gelu(x1 W1^T + b1)
  float* f2 = ff + SEQ * FFDIM;    // [512,64] ffn out
  float* x2 = f2 + SEQ * DMODEL;   // [512,64] after LN2

  embed_kernel<<<(SEQ * DMODEL) / 256, 256, 0, stream>>>(tok, pos, temb, pemb, x);

  // QKV projections: 32x4 tiles = 128 waves -> 32 blocks x 4 waves
  gemm_bias_kernel<DMODEL, DMODEL, false>
      <<<32, 128, 0, stream>>>(x, Wq, bq, q);
  gemm_bias_kernel<DMODEL, DMODEL, false>
      <<<32, 128, 0, stream>>>(x, Wk, bk, kk);
  gemm_bias_kernel<DMODEL, DMODEL, false>
      <<<32, 128, 0, stream>>>(x, Wv, bv, vv);

  // attention: one wave per (head, 16-query block) => 4*32 = 128 workgroups
  attn_kernel<<<NHEAD * (SEQ / 16), 32, 0, stream>>>(q, kk, vv, ao);

  add_ln_kernel<<<SEQ, 32, 0, stream>>>(x, ao, g1, be1, x1);

  // FFN: [512,64]x[128,64]^T (GELU) then [512,128]x[64,128]^T
  gemm_bias_kernel<FFDIM, DMODEL, true>
      <<<64, 128, 0, stream>>>(x1, W1, b1, ff);
  gemm_bias_kernel<DMODEL, FFDIM, false>
      <<<32, 128, 0, stream>>>(ff, W2, b2, f2);

  add_ln_kernel<<<SEQ, 32, 0, stream>>>(x1, f2, g2, be2, x2);

  // logits: ragged N=4 via zero-padded LDS weight tile
  logits_kernel<<<8, 128, 0, stream>>>(x2, Wo, bo, out);
}